// PiecewiseContEmbeddings_79207786873226
// MI455X (gfx1250) — compile-verified
//
#include <hip/hip_runtime.h>

typedef __attribute__((ext_vector_type(2))) float v2f;
typedef __attribute__((ext_vector_type(4))) float v4f;
typedef __attribute__((ext_vector_type(8))) float v8f;

#define B_ 4096
#define N_ 64
#define K_ 128
#define E_ 512

// ---------------------------------------------------------------------------
// Kernel 1: Wcb[n,j,:] = bias[n,:] + sum_{k<j} weight[n,k,:]
// i.e. Wcb(128x512) = L(128x128 strict-lower-triangular ones) @ W(128x512)
//                     + broadcast(bias)
// per n, computed with V_WMMA_F32_16X16X4_F32 (exact fp32 arithmetic).
// One wave produces one 16x16 tile. Triangular structure truncates the K loop
// at kb <= mb+12; the bound is readfirstlane'd so the loop is wave-uniform
// scalar control flow (EXEC untouched around the WMMA, as the ISA requires).
//
// fp32 A fragment (ISA 7.12.2): lane l, vgpr v -> M = l%16, K = v + 2*(l/16)
// fp32 B fragment (mirrored):   lane l, vgpr v -> N = l%16, K = v + 2*(l/16)
// fp32 C/D fragment:            lane l, vgpr v -> N = l%16, M = v + 8*(l/16)
// ---------------------------------------------------------------------------
__global__ __launch_bounds__(256) void prefix_wmma_kernel(const float* __restrict__ W,
                                                          const float* __restrict__ bias,
                                                          float* __restrict__ Wcb) {
  const int lane  = threadIdx.x & 31;
  const int wid   = blockIdx.x * (blockDim.x >> 5) + (threadIdx.x >> 5);
  const int ntile = wid & 31;          // 32 column tiles of 16 (E = 512)
  const int mtile = (wid >> 5) & 7;    // 8 row tiles of 16   (K = 128)
  const int n     = wid >> 8;          // 64 tables

  const int mb    = mtile << 4;
  const int nb    = ntile << 4;
  const int lhalf = lane >> 4;         // 0 or 1
  const int lmod  = lane & 15;

  const float* __restrict__ Wn = W + (size_t)n * (K_ * E_);

  v8f c = {};
  const int rowj  = mb + lmod;         // global output row covered by A's M
  const int kbmax = __builtin_amdgcn_readfirstlane(mb + 12);  // wave-uniform
  for (int kb = 0; kb <= kbmax; kb += 4) {
    const int k0 = kb + (lhalf << 1);
    v2f a, b;
    a.x = (k0     < rowj) ? 1.0f : 0.0f;   // L[j,k] = (k < j)
    a.y = (k0 + 1 < rowj) ? 1.0f : 0.0f;
    b.x = Wn[(size_t)k0       * E_ + (nb + lmod)];
    b.y = Wn[(size_t)(k0 + 1) * E_ + (nb + lmod)];
    // 8 args: (neg_a, A, neg_b, B, c_mod, C, reuse_a, reuse_b)
    c = __builtin_amdgcn_wmma_f32_16x16x4_f32(false, a, false, b,
                                              (short)0, c, false, false);
  }

  // Fold bias in: bias depends only on the column (e), which is constant
  // (nb + lmod) across all 8 accumulator VGPRs of this lane.
  const float bval = bias[(size_t)n * E_ + (nb + lmod)];

  float* __restrict__ Cn = Wcb + (size_t)n * (K_ * E_);
#pragma unroll
  for (int v = 0; v < 8; ++v) {
    const int m = mb + v + (lhalf << 3);
    Cn[(size_t)m * E_ + (nb + lmod)] = c[v] + bval;
  }
}

// ---------------------------------------------------------------------------
// Kernel 2: store-bound gather.
//   out[b,n,:] = relu(Wcb[n,bidx,:] + frac * weight[n,bidx,:])
// One wave per (b,n); wave-uniform binary search reproduces
// searchsorted(inner, x, side='left').
// Gathered rows (32 MiB of tables) are read with default temporal policy so
// they stay resident in the 192 MB L2; the 512 MiB output stream is written
// with non-temporal stores (TH=NT) so it does not evict them.
// ---------------------------------------------------------------------------
__global__ __launch_bounds__(256) void gather_relu_kernel(
    const float* __restrict__ X, const float* __restrict__ bounds,
    const float* __restrict__ W, const float* __restrict__ Wcb,
    float* __restrict__ out) {
  const int lane = threadIdx.x & 31;
  const int wid  = blockIdx.x * (blockDim.x >> 5) + (threadIdx.x >> 5);
  const int b = wid >> 6;            // / N_
  const int n = wid & (N_ - 1);

  const float x = X[(size_t)b * N_ + n];
  const float* __restrict__ brow = bounds + n * (K_ + 1);

  // bidx = first index i in inner[0..126] (= brow[1..127]) with inner[i] >= x,
  // or 127 if none. (== count of inner elements < x)
  int lo = 0, hi = K_ - 1;
  while (lo < hi) {
    const int mid = (lo + hi) >> 1;
    if (brow[1 + mid] < x) lo = mid + 1; else hi = mid;
  }
  const int bidx = lo;

  const float s    = brow[bidx];
  const float e    = brow[bidx + 1];
  const float frac = (x - s) / (e - s + 1e-8f);

  const size_t roff = (size_t)(n * K_ + bidx) * E_;
  const v4f* __restrict__ wc = (const v4f*)(Wcb + roff);
  const v4f* __restrict__ wt = (const v4f*)(W + roff);
  v4f* __restrict__ o = (v4f*)(out + ((size_t)b * N_ + n) * E_);

#pragma unroll
  for (int i = 0; i < 4; ++i) {
    const int idx = (i << 5) + lane;        // 128 float4s per row, coalesced
    const v4f a = wc[idx];
    const v4f w = wt[idx];
    v4f r;
    r.x = fmaxf(fmaf(frac, w.x, a.x), 0.0f);
    r.y = fmaxf(fmaf(frac, w.y, a.y), 0.0f);
    r.z = fmaxf(fmaf(frac, w.z, a.z), 0.0f);
    r.w = fmaxf(fmaf(frac, w.w, a.w), 0.0f);
    __builtin_nontemporal_store(r, &o[idx]);  // TH=NT: don't pollute L2
  }
}

extern "C" void kernel_launch(void* const* d_in, const int* in_sizes, int n_in,
                              void* d_out, int out_size, void* d_ws, size_t ws_size,
                              hipStream_t stream) {
  const float* X      = (const float*)d_in[0];  // (B, N)
  const float* bounds = (const float*)d_in[1];  // (N, K+1)
  const float* W      = (const float*)d_in[2];  // (N, K, E)
  const float* bias   = (const float*)d_in[3];  // (N, E)
  float*       out    = (float*)d_out;          // (B, N, E)
  float*       Wcb    = (float*)d_ws;           // N*K*E*4 = 16 MiB scratch

  // Prefix-sum (+bias) tables via WMMA: 64 * 8 * 32 = 16384 tiles, 8 waves/block.
  const int tiles = N_ * (K_ / 16) * (E_ / 16);
  prefix_wmma_kernel<<<tiles / 8, 256, 0, stream>>>(W, bias, Wcb);

  // Store-bound gather: one wave per (b, n).
  const int waves = B_ * N_;
  gather_relu_kernel<<<waves / 8, 256, 0, stream>>>(X, bounds, W, Wcb, out);
}